// SinkhornDistance_22917945491839
// MI455X (gfx1250) — compile-verified
//
#include <hip/hip_runtime.h>
#include <math.h>

#define BB 16
#define NN 4096
#define MM 512
#define CC 512
#define EPSI 0.05f
#define NITER 100
#define THRESH 0.1f

typedef float v2f __attribute__((ext_vector_type(2)));
typedef float v8f __attribute__((ext_vector_type(8)));

__device__ __forceinline__ float wave_reduce_sum(float x) {
#pragma unroll
  for (int off = 16; off > 0; off >>= 1) x += __shfl_xor(x, off, 32);
  return x;
}

// ---------------------------------------------------------------- init
__global__ void k_init(float* __restrict__ W, float* __restrict__ cost, int n) {
  int i = blockIdx.x * blockDim.x + threadIdx.x;
  if (i < n) W[i] = 0.0f;
  if (i < BB) cost[i] = 0.0f;
}

// ------------------------------------------------- reciprocal row norms
// one wave per row of CC=512 floats
__global__ void __launch_bounds__(256) k_rownorm(const float* __restrict__ src,
                                                 float* __restrict__ dst,
                                                 int nrows) {
  const int gw = (blockIdx.x * blockDim.x + threadIdx.x) >> 5;
  const int lane = threadIdx.x & 31;
  if (gw >= nrows) return;
  const float* r = src + (size_t)gw * CC;
  float s = 0.0f;
#pragma unroll
  for (int t = 0; t < 4; ++t) {
    float4 q = *(const float4*)(r + 4 * lane + 128 * t);
    s += q.x * q.x + q.y * q.y + q.z * q.z + q.w * q.w;
  }
  s = wave_reduce_sum(s);
  if (lane == 0) dst[gw] = 1.0f / fmaxf(sqrtf(s), 1e-12f);
}

// ---------------------------------------- K = exp(-(1 - s*rnx*rny)/eps)
// f32 WMMA GEMM, register-blocked: each wave computes a 32x32 tile
// (2 A-frags x 2 B-frags -> 4 accumulators), 4 loads : 4 WMMAs per k-step.
// Block = 4 waves tiling 64x64.
__global__ void __launch_bounds__(128) k_gemm_expK(
    const float* __restrict__ x, const float* __restrict__ y,
    const float* __restrict__ rnx, const float* __restrict__ rny,
    float* __restrict__ Kbuf) {
  const int wave = threadIdx.x >> 5;
  const int lane = threadIdx.x & 31;
  const int b = blockIdx.z;
  const int g = lane >> 4;   // K-half selector (K={0,1} vs K={2,3})
  const int l = lane & 15;   // row/col within fragment

  const int n_base = blockIdx.y * 64 + (wave >> 1) * 32;
  const int m_base = blockIdx.x * 64 + (wave & 1) * 32;

  const float* xp0 = x + ((size_t)b * NN + (n_base + l)) * CC + 2 * g;
  const float* xp1 = xp0 + (size_t)16 * CC;
  const float* yp0 = y + ((size_t)b * MM + (m_base + l)) * CC + 2 * g;
  const float* yp1 = yp0 + (size_t)16 * CC;

  v8f acc00 = {0.f,0.f,0.f,0.f,0.f,0.f,0.f,0.f};
  v8f acc01 = acc00, acc10 = acc00, acc11 = acc00;

#pragma unroll 4
  for (int c0 = 0; c0 < CC; c0 += 4) {
    v2f a0 = *(const v2f*)(xp0 + c0);
    v2f a1 = *(const v2f*)(xp1 + c0);
    v2f b0 = *(const v2f*)(yp0 + c0);
    v2f b1 = *(const v2f*)(yp1 + c0);
    acc00 = __builtin_amdgcn_wmma_f32_16x16x4_f32(false, a0, false, b0,
                                                  (short)0, acc00, false, false);
    acc01 = __builtin_amdgcn_wmma_f32_16x16x4_f32(false, a0, false, b1,
                                                  (short)0, acc01, false, false);
    acc10 = __builtin_amdgcn_wmma_f32_16x16x4_f32(false, a1, false, b0,
                                                  (short)0, acc10, false, false);
    acc11 = __builtin_amdgcn_wmma_f32_16x16x4_f32(false, a1, false, b1,
                                                  (short)0, acc11, false, false);
  }

  const float rm0 = rny[b * MM + m_base + l];
  const float rm1 = rny[b * MM + m_base + 16 + l];
#pragma unroll
  for (int i = 0; i < 2; ++i) {
#pragma unroll
    for (int r = 0; r < 8; ++r) {
      const int n = n_base + 16 * i + r + 8 * g;
      const float rn = rnx[b * NN + n];
      const float s0 = (i == 0) ? acc00[r] : acc10[r];
      const float s1 = (i == 0) ? acc01[r] : acc11[r];
      const size_t rowoff = ((size_t)b * NN + n) * MM;
      Kbuf[rowoff + m_base + l] =
          __expf(-(1.0f - s0 * rn * rm0) * (1.0f / EPSI));
      Kbuf[rowoff + m_base + 16 + l] =
          __expf(-(1.0f - s1 * rn * rm1) * (1.0f / EPSI));
    }
  }
}

// ------------------------------- u_new = mu / (K v + 1e-8), accumulate err
// one wave per row (b,n): 512-float dot, fully coalesced float4 loads
__global__ void __launch_bounds__(256) k_update_u(
    const float* __restrict__ Kbuf, const float* __restrict__ v,
    const float* __restrict__ u, float* __restrict__ un,
    float* __restrict__ errsum, int iter) {
  __shared__ float serr[8];
  const int wave = threadIdx.x >> 5;
  const int lane = threadIdx.x & 31;
  const int row = blockIdx.x * 8 + wave;  // in [0, B*N)
  const int b = row >> 12;                // row / 4096
  const float* kr = Kbuf + (size_t)row * MM;
  const float* vb = v + b * MM;
  float s = 0.0f;
#pragma unroll
  for (int t = 0; t < 4; ++t) {
    const int c = 4 * lane + 128 * t;
    float4 kq = *(const float4*)(kr + c);
    float4 vq = *(const float4*)(vb + c);
    s += kq.x * vq.x + kq.y * vq.y + kq.z * vq.z + kq.w * vq.w;
  }
  s = wave_reduce_sum(s);
  if (lane == 0) {
    const float unew = (1.0f / NN) / (s + 1e-8f);
    un[row] = unew;
    serr[wave] = fabsf(unew - u[row]);
  }
  __syncthreads();
  if (threadIdx.x == 0) {
    float t = 0.0f;
#pragma unroll
    for (int w = 0; w < 8; ++w) t += serr[w];
    atomicAdd(&errsum[iter], t);
  }
}

// --------------------------- partial K^T u_new: column reduction into vt
__global__ void __launch_bounds__(256) k_partial_v(
    const float* __restrict__ Kbuf, const float* __restrict__ un,
    float* __restrict__ vt) {
  __shared__ float su[256];
  const int b = blockIdx.z;
  const int n0 = blockIdx.x * 256;
  const int m = blockIdx.y * 256 + threadIdx.x;
  su[threadIdx.x] = un[b * NN + n0 + threadIdx.x];
  __syncthreads();
  const float* kp = Kbuf + ((size_t)(b * NN + n0)) * MM + m;
  float acc = 0.0f;
#pragma unroll 8
  for (int n = 0; n < 256; ++n) acc += kp[(size_t)n * MM] * su[n];
  atomicAdd(&vt[b * MM + m], acc);
}

// ------------- apply updates (freeze-aware), advance done chain, clear vt
__global__ void __launch_bounds__(256) k_finish_iter(
    const float* __restrict__ un, float* __restrict__ u,
    float* __restrict__ vt, float* __restrict__ v,
    const float* __restrict__ errsum, int* __restrict__ done, int iter) {
  const int i = blockIdx.x * 256 + threadIdx.x;
  const int dprev = done[iter];
  if (i == 0) {
    const float err = errsum[iter] * (1.0f / BB);
    done[iter + 1] = dprev | (err < THRESH ? 1 : 0);
  }
  if (!dprev && i < BB * NN) u[i] = un[i];
  if (i < BB * MM) {
    const float vn = (1.0f / MM) / (vt[i] + 1e-8f);
    if (!dprev) v[i] = vn;
    vt[i] = 0.0f;  // ready for next iteration
  }
}

// -------- pi = u*K*v (in place over K), cost[b] = sum pi * (-eps ln K)
__global__ void __launch_bounds__(256) k_final(
    float* __restrict__ KP, const float* __restrict__ u,
    const float* __restrict__ v, float* __restrict__ cost) {
  __shared__ float sred[256];
  const int b = blockIdx.y;
  const size_t base_in_b = (size_t)blockIdx.x * 2048 + (size_t)threadIdx.x * 8;
  const size_t base = (size_t)b * NN * MM + base_in_b;
  const int n = (int)(base_in_b >> 9);   // / MM, 8 elems stay in one row
  const int m0 = (int)(base_in_b & 511);
  const float ub = u[b * NN + n];
  float4 k0 = *(const float4*)(KP + base);
  float4 k1 = *(const float4*)(KP + base + 4);
  float kv[8] = {k0.x, k0.y, k0.z, k0.w, k1.x, k1.y, k1.z, k1.w};
  float pi[8];
  float local = 0.0f;
#pragma unroll
  for (int e = 0; e < 8; ++e) {
    const float p = ub * kv[e] * v[b * MM + m0 + e];
    pi[e] = p;
    const float cm = -EPSI * __logf(fmaxf(kv[e], 1e-38f));
    local += p * cm;
  }
  *(float4*)(KP + base) = make_float4(pi[0], pi[1], pi[2], pi[3]);
  *(float4*)(KP + base + 4) = make_float4(pi[4], pi[5], pi[6], pi[7]);
  sred[threadIdx.x] = local;
  __syncthreads();
  for (int off = 128; off > 0; off >>= 1) {
    if (threadIdx.x < off) sred[threadIdx.x] += sred[threadIdx.x + off];
    __syncthreads();
  }
  if (threadIdx.x == 0) atomicAdd(&cost[b], sred[0]);
}

extern "C" void kernel_launch(void* const* d_in, const int* in_sizes, int n_in,
                              void* d_out, int out_size, void* d_ws, size_t ws_size,
                              hipStream_t stream) {
  (void)in_sizes; (void)n_in; (void)out_size; (void)ws_size;
  const float* x = (const float*)d_in[0];
  const float* y = (const float*)d_in[1];
  float* cost = (float*)d_out;
  float* Kbuf = (float*)d_out + BB;  // pi region doubles as K storage

  float* W = (float*)d_ws;
  float* rnx = W;                    // B*N
  float* rny = rnx + BB * NN;        // B*M
  float* u   = rny + BB * MM;        // B*N
  float* un  = u   + BB * NN;        // B*N
  float* v   = un  + BB * NN;        // B*M
  float* vt  = v   + BB * MM;        // B*M
  float* errsum = vt + BB * MM;      // 128
  int*   done = (int*)(errsum + 128);// 128

  const int WS_FLOATS = 3 * BB * NN + 3 * BB * MM + 256;
  k_init<<<(WS_FLOATS + 255) / 256, 256, 0, stream>>>(W, cost, WS_FLOATS);

  k_rownorm<<<(BB * NN) / 8, 256, 0, stream>>>(x, rnx, BB * NN);
  k_rownorm<<<(BB * MM) / 8, 256, 0, stream>>>(y, rny, BB * MM);

  k_gemm_expK<<<dim3(MM / 64, NN / 64, BB), 128, 0, stream>>>(x, y, rnx, rny, Kbuf);

  for (int it = 0; it < NITER; ++it) {
    k_update_u<<<(BB * NN) / 8, 256, 0, stream>>>(Kbuf, v, u, un, errsum, it);
    k_partial_v<<<dim3(NN / 256, MM / 256, BB), 256, 0, stream>>>(Kbuf, un, vt);
    k_finish_iter<<<(BB * NN + 255) / 256, 256, 0, stream>>>(un, u, vt, v, errsum, done, it);
  }

  k_final<<<dim3((NN * MM) / 2048, BB), 256, 0, stream>>>(Kbuf, u, v, cost);
}